// Net_GIN_ASAP_GlobalPooling_56727928046279
// MI455X (gfx1250) — compile-verified
//
#include <hip/hip_runtime.h>
#include <math.h>

namespace {

constexpr int       B_   = 8;
constexpr int       N_   = 384;
constexpr int       NSQ_ = N_ * N_;
constexpr long long BNN_ = (long long)B_ * NSQ_;
constexpr int       BN_  = B_ * N_;
constexpr float     EPS_ = 1e-5f;
constexpr int       D0P_ = 64;     // FIN=39 padded to 64 so all GEMM dims % 64 == 0

typedef __attribute__((ext_vector_type(2))) float v2f;
typedef __attribute__((ext_vector_type(8))) float v8f;

// ---------------------------------------------------------------- utilities
__global__ void zero_kernel(float* p, long long n) {
  long long i  = (long long)blockIdx.x * blockDim.x + threadIdx.x;
  long long st = (long long)gridDim.x * blockDim.x;
  for (; i < n; i += st) p[i] = 0.f;
}

__global__ void scatter_adj_kernel(const int* __restrict__ ei, int E, float* __restrict__ adj) {
  int e = blockIdx.x * blockDim.x + threadIdx.x;
  if (e >= E) return;
  int s = ei[e];
  int t = ei[E + e];
  int b = s / N_;
  adj[(long long)b * NSQ_ + (long long)(s % N_) * N_ + (t % N_)] = 1.f;
}

__global__ void pad_x_kernel(const float* __restrict__ x, float* __restrict__ xp) {
  int row = blockIdx.x;            // 0..BN_-1
  int col = threadIdx.x;           // 0..63
  xp[(long long)row * D0P_ + col] = (col < 39) ? x[(long long)row * 39 + col] : 0.f;
}

__global__ void pad_w_kernel(const float* __restrict__ w, float* __restrict__ wp) {
  int row = blockIdx.x;            // 0..63
  for (int col = threadIdx.x; col < 128; col += blockDim.x)
    wp[row * 128 + col] = (row < 39) ? w[row * 128 + col] : 0.f;
}

__global__ void axpy_kernel(float* __restrict__ y, const float* __restrict__ x, long long n) {
  long long i  = (long long)blockIdx.x * blockDim.x + threadIdx.x;
  long long st = (long long)gridDim.x * blockDim.x;
  for (; i < n; i += st) y[i] += x[i];
}

// ------------------------------------------------ generic batched WMMA GEMM
// C[M,Nn] = op(A)[M,K] * B[K,Nn] (+bias)(ReLU)   per batch slice (z dim)
// OPA==0: A row-major [M,K]     -> A fragment pair is a contiguous float2 (b64)
// OPA==1: A stored [K,M] (A^T)  -> strided b32 loads
// Register-blocked 2x2: each wave owns a 32x32 C macro-tile (4 v8f
// accumulators), a 128-thread block covers 64x64. The second M/N sub-tile is
// addressed with a +16-float immediate offset from the same pointers, so each
// K-step is 4 v_wmma_f32_16x16x4_f32 fed by 6-8 loads and 4 pointer adds.
// Software-pipelined: fragments for step i+1 load before step i's wmmas.
// Requires M % 64 == 0, Nn % 64 == 0, K % 4 == 0 (guaranteed by padding).
template <int OPA>
__global__ __launch_bounds__(128)
void wmma_gemm_f32(const float* __restrict__ A, const float* __restrict__ Bm,
                   float* __restrict__ C, int M, int Nn, int K,
                   int lda, int ldb, int ldc,
                   long long sA, long long sB, long long sC,
                   const float* __restrict__ bias, int relu)
{
  int bz = blockIdx.z;
  const float* Ab = A  + (long long)bz * sA;
  const float* Bb = Bm + (long long)bz * sB;
  float*       Cb = C  + (long long)bz * sC;
  int wave = threadIdx.x >> 5;
  int lane = threadIdx.x & 31;
  int half = lane >> 4;
  int l16  = lane & 15;
  int m0 = blockIdx.x * 64 + (wave >> 1) * 32;   // M base of this wave's macro-tile
  int n0 = blockIdx.y * 64 + (wave & 1) * 32;    // N base
  int gm0 = m0 + l16;                            // A row, sub-tile 0 (sub-tile 1: +16)
  int gn0 = n0 + l16;                            // B/C col, sub-tile 0 (+16 for 1)

  // fragment pointers (computed once; advanced by constant byte strides)
  const float *pAa, *pAb;      // OPA=1: rows ka, ka+1 of A^T layout
  const float *pAm0, *pAm1;    // OPA=0: rows gm0, gm0+16 of A
  long long stepA;
  if (OPA) {
    pAa = Ab + (long long)(half * 2) * lda + gm0;    // [16] -> sub-tile 1
    pAb = pAa + lda;
    pAm0 = pAm1 = nullptr;
    stepA = 4LL * lda;
  } else {
    pAm0 = Ab + (long long)gm0 * lda + half * 2;     // contiguous float2
    pAm1 = pAm0 + (long long)16 * lda;
    pAa = pAb = nullptr;
    stepA = 4;
  }
  const float* pBa = Bb + (long long)(half * 2) * ldb + gn0;  // [16] -> sub-tile 1
  const float* pBb = pBa + ldb;
  const long long stepB = 4LL * ldb;

  auto load_frags = [&](v2f& a0, v2f& a1, v2f& b0, v2f& b1) {
    if (OPA) {
      a0.x = pAa[0];  a0.y = pAb[0];
      a1.x = pAa[16]; a1.y = pAb[16];
    } else {
      a0 = *(const v2f*)pAm0;          // 8B aligned (lda, k multiples of 4)
      a1 = *(const v2f*)pAm1;
    }
    b0.x = pBa[0];  b0.y = pBb[0];
    b1.x = pBa[16]; b1.y = pBb[16];
  };

  v8f acc00 = {}, acc01 = {}, acc10 = {}, acc11 = {};
  v2f a0, a1, b0, b1;
  load_frags(a0, a1, b0, b1);

  int iters = K >> 2;
#pragma unroll 2
  for (int it = 0; it < iters - 1; ++it) {
    if (OPA) { pAa += stepA; pAb += stepA; }
    else     { pAm0 += stepA; pAm1 += stepA; }
    pBa += stepB; pBb += stepB;
    v2f na0, na1, nb0, nb1;
    load_frags(na0, na1, nb0, nb1);            // prefetch next K-step
    acc00 = __builtin_amdgcn_wmma_f32_16x16x4_f32(false, a0, false, b0, (short)0, acc00, false, false);
    acc01 = __builtin_amdgcn_wmma_f32_16x16x4_f32(false, a0, false, b1, (short)0, acc01, false, false);
    acc10 = __builtin_amdgcn_wmma_f32_16x16x4_f32(false, a1, false, b0, (short)0, acc10, false, false);
    acc11 = __builtin_amdgcn_wmma_f32_16x16x4_f32(false, a1, false, b1, (short)0, acc11, false, false);
    a0 = na0; a1 = na1; b0 = nb0; b1 = nb1;
  }
  acc00 = __builtin_amdgcn_wmma_f32_16x16x4_f32(false, a0, false, b0, (short)0, acc00, false, false);
  acc01 = __builtin_amdgcn_wmma_f32_16x16x4_f32(false, a0, false, b1, (short)0, acc01, false, false);
  acc10 = __builtin_amdgcn_wmma_f32_16x16x4_f32(false, a1, false, b0, (short)0, acc10, false, false);
  acc11 = __builtin_amdgcn_wmma_f32_16x16x4_f32(false, a1, false, b1, (short)0, acc11, false, false);

  float bn0 = bias ? bias[gn0]      : 0.f;
  float bn1 = bias ? bias[gn0 + 16] : 0.f;
#pragma unroll
  for (int i = 0; i < 8; ++i) {               // C: vgpr i -> row i (+8 hi half)
    int r0 = m0 + i + half * 8;
    int r1 = r0 + 16;
    float v;
    v = acc00[i] + bn0; if (relu) v = fmaxf(v, 0.f); Cb[(long long)r0 * ldc + gn0]      = v;
    v = acc01[i] + bn1; if (relu) v = fmaxf(v, 0.f); Cb[(long long)r0 * ldc + gn0 + 16] = v;
    v = acc10[i] + bn0; if (relu) v = fmaxf(v, 0.f); Cb[(long long)r1 * ldc + gn0]      = v;
    v = acc11[i] + bn1; if (relu) v = fmaxf(v, 0.f); Cb[(long long)r1 * ldc + gn0 + 16] = v;
  }
  (void)M;
}

// -------------------------------------------------------------- batch norm
__global__ __launch_bounds__(256)
void bn_stats_kernel(const float* __restrict__ H, int ld, int nreal,
                     float* __restrict__ meanv, float* __restrict__ varv)
{
  int j = blockIdx.x;                        // feature
  __shared__ float ss[256], sq[256];
  float s = 0.f, q = 0.f;
  int total = B_ * nreal;
  for (int r = threadIdx.x; r < total; r += 256) {
    int b = r / nreal, t = r % nreal;
    float v = H[((long long)b * N_ + t) * ld + j];
    s += v; q += v * v;
  }
  ss[threadIdx.x] = s; sq[threadIdx.x] = q;
  __syncthreads();
  for (int o = 128; o > 0; o >>= 1) {
    if (threadIdx.x < o) { ss[threadIdx.x] += ss[threadIdx.x + o];
                           sq[threadIdx.x] += sq[threadIdx.x + o]; }
    __syncthreads();
  }
  if (threadIdx.x == 0) {
    float m = ss[0] / (float)total;
    meanv[j] = m;
    varv[j]  = sq[0] / (float)total - m * m;
  }
}

__global__ void bn_apply_relu_kernel(float* __restrict__ H, int d, long long n,
                                     const float* __restrict__ g, const float* __restrict__ be,
                                     const float* __restrict__ m, const float* __restrict__ v)
{
  long long i  = (long long)blockIdx.x * blockDim.x + threadIdx.x;
  long long st = (long long)gridDim.x * blockDim.x;
  for (; i < n; i += st) {
    int j = (int)(i % d);
    float h = (H[i] - m[j]) * rsqrtf(v[j] + EPS_) * g[j] + be[j];
    H[i] = fmaxf(h, 0.f);
  }
}

// ---------------------------------------------------------------- ASAP ops
__global__ void build_asl_kernel(const float* __restrict__ A, int nreal, float* __restrict__ ASL) {
  long long i  = (long long)blockIdx.x * blockDim.x + threadIdx.x;
  long long st = (long long)gridDim.x * blockDim.x;
  for (; i < BNN_; i += st) {
    long long r = i % NSQ_;
    int s = (int)(r / N_), t = (int)(r % N_);
    float v = A[i];
    if (s == t && t < nreal) v = 1.f;        // add_remaining_self_loops(fill=1)
    ASL[i] = v;
  }
}

// x_q[t,d] = max over sources s with a_sl[s,t] of x[s,d]
__global__ void xq_max_kernel(const float* __restrict__ ASL, const float* __restrict__ X,
                              float* __restrict__ XQ, int d)
{
  int b = blockIdx.z, t = blockIdx.x, j = threadIdx.x;
  if (j >= d) return;
  const float* aslb = ASL + (long long)b * NSQ_;
  const float* xb   = X   + (long long)b * N_ * d;
  float best = -1e30f; int any = 0;
  for (int s = 0; s < N_; ++s) {
    if (aslb[(long long)s * N_ + t] != 0.f) {
      any = 1;
      best = fmaxf(best, xb[(long long)s * d + j]);
    }
  }
  XQ[((long long)b * N_ + t) * d + j] = any ? best : 0.f;
}

__global__ void qr_dot_kernel(const float* __restrict__ XQL, const float* __restrict__ X,
                              const float* __restrict__ attw, int d,
                              float* __restrict__ Q, float* __restrict__ R)
{
  int i = blockIdx.x * blockDim.x + threadIdx.x;   // b*N_+node
  if (i >= BN_) return;
  float q = 0.f, r = 0.f;
  for (int j = 0; j < d; ++j) {
    q += XQL[(long long)i * d + j] * attw[j];
    r += X  [(long long)i * d + j] * attw[d + j];
  }
  Q[i] = q; R[i] = r;
}

// per target column t: masked leaky-relu softmax over sources s; also deg[t]
__global__ __launch_bounds__(128)
void smat_softmax_kernel(const float* __restrict__ ASL, const float* __restrict__ Q,
                         const float* __restrict__ R, const float* __restrict__ attb,
                         float* __restrict__ SMAT, float* __restrict__ DEG)
{
  int b = blockIdx.z, t = blockIdx.x;
  const float* aslb = ASL + (long long)b * NSQ_;
  float qt = Q[b * N_ + t] + attb[0];
  float sc[3]; int mk[3];
  float mx = -3.4e38f, cnt = 0.f;
#pragma unroll
  for (int c = 0; c < 3; ++c) {
    int s = threadIdx.x + c * 128;
    mk[c] = (aslb[(long long)s * N_ + t] != 0.f);
    float v = R[b * N_ + s] + qt;
    v = (v >= 0.f) ? v : 0.2f * v;           // leaky_relu slope 0.2
    sc[c] = v;
    if (mk[c]) { mx = fmaxf(mx, v); cnt += 1.f; }
  }
  __shared__ float red[128];
  red[threadIdx.x] = mx; __syncthreads();
  for (int o = 64; o > 0; o >>= 1) {
    if (threadIdx.x < o) red[threadIdx.x] = fmaxf(red[threadIdx.x], red[threadIdx.x + o]);
    __syncthreads();
  }
  mx = red[0]; __syncthreads();
  float sum = 0.f;
#pragma unroll
  for (int c = 0; c < 3; ++c) if (mk[c]) sum += __expf(sc[c] - mx);
  red[threadIdx.x] = sum; __syncthreads();
  for (int o = 64; o > 0; o >>= 1) {
    if (threadIdx.x < o) red[threadIdx.x] += red[threadIdx.x + o];
    __syncthreads();
  }
  sum = red[0]; __syncthreads();
  red[threadIdx.x] = cnt; __syncthreads();
  for (int o = 64; o > 0; o >>= 1) {
    if (threadIdx.x < o) red[threadIdx.x] += red[threadIdx.x + o];
    __syncthreads();
  }
  float deg = red[0];
  float inv = (sum > 0.f) ? 1.f / sum : 0.f;
#pragma unroll
  for (int c = 0; c < 3; ++c) {
    int s = threadIdx.x + c * 128;
    SMAT[(long long)b * NSQ_ + (long long)s * N_ + t] =
        mk[c] ? __expf(sc[c] - mx) * inv : 0.f;
  }
  if (threadIdx.x == 0) DEG[b * N_ + t] = deg;
}

__global__ void le_dots_kernel(const float* __restrict__ XNEW, int d,
                               const float* __restrict__ w1, const float* __restrict__ b1,
                               const float* __restrict__ w2,
                               const float* __restrict__ w3, const float* __restrict__ b3,
                               float* __restrict__ L1, float* __restrict__ L2, float* __restrict__ L3)
{
  int i = blockIdx.x * blockDim.x + threadIdx.x;
  if (i >= BN_) return;
  float a = 0.f, c = 0.f, e = 0.f;
  for (int j = 0; j < d; ++j) {
    float x = XNEW[(long long)i * d + j];
    a += x * w1[j]; c += x * w2[j]; e += x * w3[j];
  }
  L1[i] = a + b1[0]; L2[i] = c; L3[i] = e + b3[0];
}

__global__ void fitness_kernel(const float* __restrict__ ASL,
                               const float* __restrict__ L1, const float* __restrict__ L2,
                               const float* __restrict__ L3, const float* __restrict__ DEG,
                               float* __restrict__ FIT)
{
  int b = blockIdx.z;
  int t = blockIdx.x * blockDim.x + threadIdx.x;
  if (t >= N_) return;
  const float* aslb = ASL + (long long)b * NSQ_;
  float s = 0.f;
  for (int q = 0; q < N_; ++q) s += aslb[(long long)q * N_ + t] * L1[b * N_ + q];
  float z = L3[b * N_ + t] + s - DEG[b * N_ + t] * L2[b * N_ + t];
  FIT[b * N_ + t] = 1.f / (1.f + __expf(-z));
}

// exact stable top-k by ranking (ties broken by lower index, like lax.top_k)
__global__ __launch_bounds__(384)
void topk_kernel(const float* __restrict__ FIT, int nreal, int k,
                 int* __restrict__ IDX, float* __restrict__ VALS)
{
  int b = blockIdx.x, t = threadIdx.x;
  if (t >= nreal) return;
  float ft = FIT[b * N_ + t];
  int rank = 0;
  for (int s = 0; s < nreal; ++s) {
    float fs = FIT[b * N_ + s];
    rank += (fs > ft) || (fs == ft && s < t);
  }
  if (rank < k) { IDX[b * N_ + rank] = t; VALS[b * N_ + rank] = ft; }
}

__global__ void select_x_kernel(const float* __restrict__ XNEW, const int* __restrict__ IDX,
                                const float* __restrict__ VALS, int k, int d,
                                float* __restrict__ XSEL)
{
  int b = blockIdx.z, p = blockIdx.x, j = threadIdx.x;
  if (j >= d) return;
  float v = 0.f;
  if (p < k) {
    int i = IDX[b * N_ + p];
    v = XNEW[((long long)b * N_ + i) * d + j] * VALS[b * N_ + p];
  }
  XSEL[((long long)b * N_ + p) * d + j] = v;   // zero-pad rows p>=k
}

__global__ __launch_bounds__(128)
void select_s_kernel(const float* __restrict__ SMAT, const int* __restrict__ IDX,
                     int k, float* __restrict__ SSEL)
{
  int b = blockIdx.z, p = blockIdx.x;
  int idx = (p < k) ? IDX[b * N_ + p] : 0;
  for (int s = threadIdx.x; s < N_; s += 128) {
    float v = (p < k) ? SMAT[(long long)b * NSQ_ + (long long)s * N_ + idx] : 0.f;
    SSEL[(long long)b * NSQ_ + (long long)s * N_ + p] = v;
  }
}

__global__ void readout_sum_kernel(const float* __restrict__ XSEL, int d,
                                   float* __restrict__ RO, int off)
{
  int b = blockIdx.z;
  int j = blockIdx.x * blockDim.x + threadIdx.x;
  if (j >= d) return;
  float s = 0.f;
  for (int p = 0; p < N_; ++p) s += XSEL[((long long)b * N_ + p) * d + j];
  RO[b * 512 + off + j] = s;
}

__global__ void binarize_adj_kernel(const float* __restrict__ RAW, int k, float* __restrict__ AN) {
  long long i  = (long long)blockIdx.x * blockDim.x + threadIdx.x;
  long long st = (long long)gridDim.x * blockDim.x;
  for (; i < BNN_; i += st) {
    long long r = i % NSQ_;
    int p = (int)(r / N_), q = (int)(r % N_);
    AN[i] = (p < k && q < k && p != q && RAW[i] != 0.f) ? 1.f : 0.f;
  }
}

__global__ __launch_bounds__(256)
void fc_head_kernel(const float* __restrict__ RO,
                    const float* __restrict__ w1, const float* __restrict__ b1,
                    const float* __restrict__ w2, const float* __restrict__ b2,
                    float* __restrict__ out)
{
  int b = blockIdx.x;
  __shared__ float hid[256];
  int j = threadIdx.x;
  float acc = b1[j];
  for (int i = 0; i < 512; ++i) acc += RO[b * 512 + i] * w1[i * 256 + j];
  hid[j] = acc;
  __syncthreads();
  if (j < 30) {
    float o = b2[j];
    for (int i = 0; i < 256; ++i) o += hid[i] * w2[i * 30 + j];
    out[b * 30 + j] = o;
  }
}

} // namespace

extern "C" void kernel_launch(void* const* d_in, const int* in_sizes, int n_in,
                              void* d_out, int out_size, void* d_ws, size_t ws_size,
                              hipStream_t stream)
{
  const float* x  = (const float*)d_in[0];
  const int*   ei = (const int*)d_in[1];
  auto F = [&](int i) { return (const float*)d_in[i]; };
  const int E = in_sizes[1] / 2;

  // -------- workspace layout (floats) --------
  float* ws = (float*)d_ws;  long long o = 0;
  float* ADJ_A = ws + o; o += BNN_;
  float* ADJ_B = ws + o; o += BNN_;
  float* ASL   = ws + o; o += BNN_;
  float* SMAT  = ws + o; o += BNN_;   // also reused for raw coarsened adjacency
  float* SSEL  = ws + o; o += BNN_;
  float* TMPNN = ws + o; o += BNN_;
  float* XPAD  = ws + o; o += (long long)BN_ * D0P_;
  float* WPAD  = ws + o; o += D0P_ * 128;
  float* AGG   = ws + o; o += (long long)BN_ * 256;
  float* HTMP  = ws + o; o += (long long)BN_ * 256;
  float* XCUR  = ws + o; o += (long long)BN_ * 256;
  float* XQ    = ws + o; o += (long long)BN_ * 256;
  float* XQL   = ws + o; o += (long long)BN_ * 256;
  float* XNEW  = ws + o; o += (long long)BN_ * 256;
  float* XS0   = ws + o; o += (long long)BN_ * 256;
  float* XS1   = ws + o; o += (long long)BN_ * 256;
  float* Qv    = ws + o; o += BN_;
  float* Rv    = ws + o; o += BN_;
  float* DEG   = ws + o; o += BN_;
  float* L1    = ws + o; o += BN_;
  float* L2    = ws + o; o += BN_;
  float* L3    = ws + o; o += BN_;
  float* FIT   = ws + o; o += BN_;
  float* VALS  = ws + o; o += BN_;
  float* MEANV = ws + o; o += 256;
  float* VARV  = ws + o; o += 256;
  float* RO    = ws + o; o += B_ * 512;
  int*   IDX   = (int*)(ws + o); o += BN_;
  (void)ws_size; (void)n_in;

  auto gemm = [&](const float* A, const float* Bm, float* C, int M, int Nn, int K,
                  int lda, int ldb, int ldc, long long sA, long long sB, long long sC,
                  const float* bias, int relu, int opA) {
    dim3 g(M / 64, Nn / 64, B_), blk(128);
    if (opA)
      wmma_gemm_f32<1><<<g, blk, 0, stream>>>(A, Bm, C, M, Nn, K, lda, ldb, ldc,
                                              sA, sB, sC, bias, relu);
    else
      wmma_gemm_f32<0><<<g, blk, 0, stream>>>(A, Bm, C, M, Nn, K, lda, ldb, ldc,
                                              sA, sB, sC, bias, relu);
  };

  // -------- setup --------
  zero_kernel<<<2048, 256, 0, stream>>>(ADJ_A, BNN_);
  scatter_adj_kernel<<<(E + 255) / 256, 256, 0, stream>>>(ei, E, ADJ_A);
  pad_x_kernel<<<BN_, D0P_, 0, stream>>>(x, XPAD);
  pad_w_kernel<<<D0P_, 128, 0, stream>>>(F(3), WPAD);

  // per-level config
  const int pbase[3]  = {3, 18, 33};
  const int dinps[3]  = {D0P_, 128, 128};
  const int douts[3]  = {128, 128, 256};
  const int nreals[3] = {384, 269, 189};
  const int kk[3]     = {269, 189, 133};
  const int rooff[3]  = {0, 128, 256};
  float* Acurs[3]  = {ADJ_A, ADJ_B, ADJ_A};
  float* Anexts[3] = {ADJ_B, ADJ_A, ADJ_B};
  const float* Xins[3] = {XPAD, XS0, XS1};
  float* XSouts[3] = {XS0, XS1, XS0};

  for (int L = 0; L < 3; ++L) {
    const int pb = pbase[L];
    const int dinp = dinps[L], dout = douts[L], nreal = nreals[L], k = kk[L];
    const float* Xin = Xins[L];
    float* Acur = Acurs[L];
    const float* W1 = (L == 0) ? WPAD : F(pb + 0);
    long long sX = (long long)N_ * dinp, sD = (long long)N_ * dout;

    // ---- GIN block ----
    // agg = adj^T @ x
    gemm(Acur, Xin, AGG, N_, dinp, N_, N_, dinp, dinp, NSQ_, sX, sX, nullptr, 0, 1);
    axpy_kernel<<<1024, 256, 0, stream>>>(AGG, Xin, (long long)B_ * sX);     // x + agg
    // h = (x+agg) @ W1 + b1
    gemm(AGG, W1, HTMP, N_, dout, dinp, dinp, dout, dout, sX, 0, sD, F(pb + 1), 0, 0);
    bn_stats_kernel<<<dout, 256, 0, stream>>>(HTMP, dout, nreal, MEANV, VARV);
    bn_apply_relu_kernel<<<1024, 256, 0, stream>>>(HTMP, dout, (long long)B_ * sD,
                                                   F(pb + 2), F(pb + 3), MEANV, VARV);
    // x = relu(h_bn_relu @ W2 + b2)
    gemm(HTMP, F(pb + 4), XCUR, N_, dout, dout, dout, dout, dout, sD, 0, sD, F(pb + 5), 1, 0);

    // ---- ASAP pooling ----
    build_asl_kernel<<<2048, 256, 0, stream>>>(Acur, nreal, ASL);
    xq_max_kernel<<<dim3(N_, 1, B_), dout, 0, stream>>>(ASL, XCUR, XQ, dout);
    gemm(XQ, F(pb + 6), XQL, N_, dout, dout, dout, dout, dout, sD, 0, sD, F(pb + 7), 0, 0);
    qr_dot_kernel<<<(BN_ + 255) / 256, 256, 0, stream>>>(XQL, XCUR, F(pb + 8), dout, Qv, Rv);
    smat_softmax_kernel<<<dim3(N_, 1, B_), 128, 0, stream>>>(ASL, Qv, Rv, F(pb + 9), SMAT, DEG);
    // x_new = s_mat^T @ x
    gemm(SMAT, XCUR, XNEW, N_, dout, N_, N_, dout, dout, NSQ_, sD, sD, nullptr, 0, 1);
    le_dots_kernel<<<(BN_ + 255) / 256, 256, 0, stream>>>(XNEW, dout, F(pb + 10), F(pb + 11),
                                                          F(pb + 12), F(pb + 13), F(pb + 14),
                                                          L1, L2, L3);
    fitness_kernel<<<dim3(3, 1, B_), 128, 0, stream>>>(ASL, L1, L2, L3, DEG, FIT);
    topk_kernel<<<B_, 384, 0, stream>>>(FIT, nreal, k, IDX, VALS);
    select_x_kernel<<<dim3(N_, 1, B_), dout, 0, stream>>>(XNEW, IDX, VALS, k, dout, XSouts[L]);
    select_s_kernel<<<dim3(N_, 1, B_), 128, 0, stream>>>(SMAT, IDX, k, SSEL);
    readout_sum_kernel<<<dim3((dout + 127) / 128, 1, B_), 128, 0, stream>>>(
        XSouts[L], dout, RO, rooff[L]);
    // A' = (S_sel^T @ A_sl) @ S_sel ; remove diag; binarize
    gemm(SSEL, ASL, TMPNN, N_, N_, N_, N_, N_, N_, NSQ_, NSQ_, NSQ_, nullptr, 0, 1);
    gemm(TMPNN, SSEL, SMAT, N_, N_, N_, N_, N_, N_, NSQ_, NSQ_, NSQ_, nullptr, 0, 0);
    binarize_adj_kernel<<<2048, 256, 0, stream>>>(SMAT, k, Anexts[L]);
  }

  // ---- FC head: out = fc2(fc1(readout)) ----
  fc_head_kernel<<<B_, 256, 0, stream>>>(RO, F(48), F(49), F(50), F(51), (float*)d_out);
  (void)out_size;
}